// MemoryInsDis_88227218195015
// MI455X (gfx1250) — compile-verified
//
#include <hip/hip_runtime.h>
#include <cstdint>

typedef float v2f __attribute__((ext_vector_type(2)));
typedef float v8f __attribute__((ext_vector_type(8)));

constexpr int BS    = 256;      // bs
constexpr int D     = 128;      // embedding dim
constexpr int K     = 4096;     // noise samples per row
constexpr int BATCH = 64;       // batch_size
constexpr int CLIPS = 4;
constexpr int NLEN  = 1000000;
constexpr int COLS  = K + 1;    // 4097
constexpr size_t OUTS_ELEMS = (size_t)BS * COLS;   // 1,048,832
constexpr size_t PROBS_OFF  = OUTS_ELEMS;
constexpr size_t NEWMEM_OFF = OUTS_ELEMS + 1;

// ---- positive logits: mean dot with the 3 sibling clips -> exp(./T) ----
__global__ void pos_kernel(const float* __restrict__ x,
                           const float* __restrict__ params,
                           float* __restrict__ outs,
                           float* __restrict__ wsum) {
    int i = threadIdx.x;
    if (i >= BS) return;
    int v = i % BATCH;
    float dot = 0.f;
    for (int c = 0; c < CLIPS; ++c) {
        int sib = v + c * BATCH;
        if (sib == i) continue;
        const float* xa = x + (size_t)i * D;
        const float* xb = x + (size_t)sib * D;
        float s = 0.f;
        #pragma unroll 8
        for (int t = 0; t < D; ++t) s += xa[t] * xb[t];
        dot += s;
    }
    dot *= (1.0f / (CLIPS - 1));
    float e = __expf(dot / params[1]);
    outs[(size_t)i * COLS] = e;
    atomicAdd(&wsum[0], e);
    atomicAdd(&wsum[1 + i], e);
}

// ---- negative logits via V_WMMA_F32_16X16X4_F32 over gathered rows ----
// One wave handles 16 noise rows for one b.  A = 16x4 tile of gathered
// memory rows (lane l<16: K=0..1, lane l+16: K=2..3, per ISA layout);
// B broadcasts the x[b] K-chunk across all 16 N columns, so D[:, n] holds
// the 16 dot products in every column.  Accumulate 32 steps of K=4.
__global__ void __launch_bounds__(256)
neg_kernel(const float* __restrict__ x,
           const float* __restrict__ memory,
           const float* __restrict__ params,
           const int* __restrict__ noise_idx,
           float* __restrict__ outs,
           float* __restrict__ wsum) {
    __shared__ float xs[D];
    __shared__ float bsum;
    const int b   = blockIdx.y;
    const int tid = threadIdx.x;
    if (tid < D) xs[tid] = x[(size_t)b * D + tid];
    if (tid == 0) bsum = 0.f;
    __syncthreads();

    const int lane = tid & 31;
    const int wave = tid >> 5;
    const int t0   = (blockIdx.x * 8 + wave) * 16;   // 16 noise idx per wave
    const int m    = lane & 15;
    const int c    = (lane >> 4) * 2;                // K sub-offset: 0 or 2
    const int row  = noise_idx[(size_t)b * K + t0 + m];
    const float* arow = memory + (size_t)row * D + c;

    v8f acc = {0.f, 0.f, 0.f, 0.f, 0.f, 0.f, 0.f, 0.f};
    #pragma unroll 4
    for (int it = 0; it < D / 4; ++it) {
        v2f a;  a.x  = arow[4 * it];     a.y  = arow[4 * it + 1];
        v2f bb; bb.x = xs[4 * it + c];   bb.y = xs[4 * it + c + 1];
        acc = __builtin_amdgcn_wmma_f32_16x16x4_f32(
                  false, a, false, bb, (short)0, acc, false, false);
    }

    const float invT = 1.0f / params[1];
    if ((lane & 15) == 0) {                 // lanes 0 and 16 hold column N=0
        const int m0 = (lane >> 4) * 8;     // lane 0: M=0..7, lane 16: M=8..15
        float partial = 0.f;
        size_t base = (size_t)b * COLS + 1 + (size_t)(t0 + m0);
        #pragma unroll
        for (int r = 0; r < 8; ++r) {
            float e = __expf(acc[r] * invT);
            outs[base + r] = e;
            partial += e;
        }
        atomicAdd(&bsum, partial);
    }
    __syncthreads();
    if (tid == 0) {
        atomicAdd(&wsum[0], bsum);          // global sum for Z
        atomicAdd(&wsum[1 + b], bsum);      // row sum for probs
    }
}

// ---- probs = mean_b( e[b,0] / rowsum_b ); Z-invariant, uses unscaled e ----
__global__ void probs_kernel(const float* __restrict__ outs,
                             const float* __restrict__ wsum,
                             float* __restrict__ probs_out) {
    __shared__ float red[BS];
    int t = threadIdx.x;
    red[t] = outs[(size_t)t * COLS] / wsum[1 + t];
    __syncthreads();
    for (int s = BS / 2; s > 0; s >>= 1) {
        if (t < s) red[t] += red[t + s];
        __syncthreads();
    }
    if (t == 0) probs_out[0] = red[0] / (float)BS;
}

// ---- outs /= Z, with Z = Zmom * (mean(e)*nLen) + (1-Zmom) * Z0 ----
__global__ void scale_kernel(float* __restrict__ outs,
                             const float* __restrict__ wsum,
                             const float* __restrict__ params) {
    float S    = wsum[0];
    float Zm   = params[2];
    float Znew = (S / (float)OUTS_ELEMS) * (float)NLEN;
    float inv  = 1.0f / (Zm * Znew + (1.0f - Zm) * params[0]);
    size_t i      = (size_t)blockIdx.x * blockDim.x + threadIdx.x;
    size_t stride = (size_t)gridDim.x * blockDim.x;
    for (; i < OUTS_ELEMS; i += stride) outs[i] *= inv;
}

// ---- bulk copy memory -> new_memory (dst slice is only 4B-aligned) ----
__global__ void copy_kernel(const float* __restrict__ src,
                            float* __restrict__ dst, size_t n) {
    size_t i      = (size_t)blockIdx.x * blockDim.x + threadIdx.x;
    size_t stride = (size_t)gridDim.x * blockDim.x;
    for (; i < n; i += stride) dst[i] = src[i];
}

// ---- scatter-update 64 rows: mm*mem[idx] + (1-mm)*mean(clips), renorm ----
__global__ void update_kernel(const float* __restrict__ x,
                              const float* __restrict__ memory,
                              const int* __restrict__ idxs,
                              const float* __restrict__ params,
                              float* __restrict__ newmem) {
    __shared__ float red[D];
    int v = blockIdx.x;         // video 0..63
    int t = threadIdx.x;        // dim 0..127
    int idx = idxs[v];
    float mm = params[3];
    float xm = 0.f;
    #pragma unroll
    for (int c = 0; c < CLIPS; ++c) xm += x[(size_t)(v + c * BATCH) * D + t];
    xm *= (1.0f / CLIPS);
    float nv = mm * memory[(size_t)idx * D + t] + (1.0f - mm) * xm;
    red[t] = nv * nv;
    __syncthreads();
    for (int s = D / 2; s > 0; s >>= 1) {
        if (t < s) red[t] += red[t + s];
        __syncthreads();
    }
    newmem[(size_t)idx * D + t] = nv * rsqrtf(red[0]);
}

extern "C" void kernel_launch(void* const* d_in, const int* in_sizes, int n_in,
                              void* d_out, int out_size, void* d_ws, size_t ws_size,
                              hipStream_t stream) {
    const float* x         = (const float*)d_in[0];
    const float* memory    = (const float*)d_in[1];
    const float* params    = (const float*)d_in[2];
    const int*   idxs      = (const int*)d_in[3];
    const int*   noise_idx = (const int*)d_in[4];
    // d_in[5] = i, d_in[6] = clips_num: fixed by problem shape

    float* out  = (float*)d_out;
    float* wsum = (float*)d_ws;   // [0]=global sum, [1..256]=row sums

    hipMemsetAsync(d_ws, 0, (1 + BS) * sizeof(float), stream);

    pos_kernel<<<1, BS, 0, stream>>>(x, params, out, wsum);

    dim3 ngrid(K / 128, BS);      // 32 x 256 blocks, 8 waves * 16 rows each
    neg_kernel<<<ngrid, 256, 0, stream>>>(x, memory, params, noise_idx, out, wsum);

    probs_kernel<<<1, BS, 0, stream>>>(out, wsum, out + PROBS_OFF);
    scale_kernel<<<1024, 256, 0, stream>>>(out, wsum, params);

    copy_kernel<<<4096, 256, 0, stream>>>(memory, out + NEWMEM_OFF, (size_t)NLEN * D);
    update_kernel<<<BATCH, D, 0, stream>>>(x, memory, idxs, params, out + NEWMEM_OFF);
}